// GINBlock_1365799600617
// MI455X (gfx1250) — compile-verified
//
#include <hip/hip_runtime.h>

typedef __attribute__((ext_vector_type(2))) float v2f;
typedef __attribute__((ext_vector_type(8))) float v8f;

#define BB    4
#define C     128
#define NTOK  4096
#define KNN   9
#define TILE  128
#define LDA   132                 // padded stride for the d2 tile (floats)
#define SP    288                 // interleaved pair-row stride: 2*TILE+32 (==32 mod 64)
#define NB    (NTOK / TILE)       // 32 column tiles

// interleaved-pair LDS index: element (c, j) of a [C][TILE] c-major matrix
__device__ __forceinline__ int ilx(int c, int j) {
    return (c >> 1) * SP + j * 2 + (c & 1);
}

// ---------------------------------------------------------------- init
__global__ __launch_bounds__(256) void k_init(float* sums, float* sumsq) {
    int t = threadIdx.x;
    if (t < C) { sums[t] = 0.0f; sumsq[t] = 0.0f; }
}

// ------------------------------------------------------- squared norms
__global__ __launch_bounds__(256) void k_sqnorm(const float* __restrict__ x,
                                                float* __restrict__ sq) {
    int gid = blockIdx.x * 256 + threadIdx.x;        // over B*N
    int b = gid / NTOK, n = gid % NTOK;
    const float* xb = x + (size_t)b * C * NTOK;
    float s = 0.0f;
    for (int c = 0; c < C; ++c) {
        float v = xb[(size_t)c * NTOK + n];
        s += v * v;
    }
    sq[gid] = s;
}

// ------------------------------------------- Gram (f32 WMMA) + top-9
__global__ __launch_bounds__(256) void k_gram_topk(const float* __restrict__ x,
                                                   const float* __restrict__ sq,
                                                   int* __restrict__ knn) {
    extern __shared__ char smem[];
    float* lds_a   = (float*)smem;                 // (C/2)*SP  row-block, interleaved
    float* lds_b   = lds_a + (C / 2) * SP;         // (C/2)*SP  col-block, interleaved
    float* lds_d   = lds_b + (C / 2) * SP;         // [TILE][LDA] d2 tile
    float* lds_sqr = lds_d + TILE * LDA;           // [TILE]
    float* lds_sqc = lds_sqr + TILE;               // [TILE]

    const int tid  = threadIdx.x;
    const int b    = blockIdx.x / NB;
    const int n0   = (blockIdx.x % NB) * TILE;
    const float* xb  = x  + (size_t)b * C * NTOK;
    const float* sqb = sq + (size_t)b * NTOK;

    for (int i = tid; i < C * TILE; i += 256) {
        int c = i >> 7, t = i & 127;
        lds_a[ilx(c, t)] = xb[(size_t)c * NTOK + n0 + t];
    }
    if (tid < TILE) lds_sqr[tid] = sqb[n0 + tid];

    float tv[KNN]; int ti[KNN];
#pragma unroll
    for (int j = 0; j < KNN; ++j) { tv[j] = 3.4e38f; ti[j] = 0; }

    const int lane = tid & 31;
    const int wv   = tid >> 5;        // 0..7 -> mi band
    const int half = lane >> 4;       // K-half selector
    const int l16  = lane & 15;
    const int mrow = wv * 16 + l16;

    for (int mt = 0; mt < NB; ++mt) {
        const int m0 = mt * TILE;
        __syncthreads();                               // prev scan done
        for (int i = tid; i < C * TILE; i += 256) {
            int c = i >> 7, t = i & 127;
            lds_b[ilx(c, t)] = xb[(size_t)c * NTOK + m0 + t];
        }
        if (tid < TILE) lds_sqc[tid] = sqb[m0 + tid];
        __syncthreads();

        // prefetch next column tile into cache while we compute (512 lines / 256 thr)
        {
            const int mn = ((mt + 1 < NB) ? (mt + 1) : mt) * TILE;
#pragma unroll
            for (int p = 0; p < 2; ++p) {
                int L = tid + p * 256;                 // cacheline id 0..511
                int c = L >> 2, t = (L & 3) * 32;
                __builtin_prefetch(&xb[(size_t)c * NTOK + mn + t], 0, 0);
            }
        }

        // wave wv computes a 16x128 band: 8 accumulators, K=128 in steps of 4
        v8f acc[8];
#pragma unroll
        for (int ni = 0; ni < 8; ++ni)
            acc[ni] = (v8f){0,0,0,0,0,0,0,0};

        for (int c0 = 0; c0 < C; c0 += 4) {
            const int kp = (c0 >> 1) + half;           // pair-row index
            v2f a = *(const v2f*)&lds_a[kp * SP + mrow * 2];
#pragma unroll
            for (int ni = 0; ni < 8; ++ni) {
                v2f bf = *(const v2f*)&lds_b[kp * SP + (ni * 16 + l16) * 2];
                acc[ni] = __builtin_amdgcn_wmma_f32_16x16x4_f32(
                    false, a, false, bf, (short)0, acc[ni], false, false);
            }
        }
        // d2 = sq[n] + sq[m] - 2G
#pragma unroll
        for (int ni = 0; ni < 8; ++ni) {
            int col = ni * 16 + l16;
#pragma unroll
            for (int r = 0; r < 8; ++r) {
                int row = wv * 16 + half * 8 + r;
                lds_d[row * LDA + col] =
                    lds_sqr[row] + lds_sqc[col] - 2.0f * acc[ni][r];
            }
        }
        __syncthreads();

        // top-9 maintenance: one thread per row, registers only
        if (tid < TILE) {
            const float* drow = &lds_d[tid * LDA];
            for (int ml = 0; ml < TILE; ++ml) {
                float v = drow[ml];
                if (v < tv[KNN - 1]) {
                    tv[KNN - 1] = v; ti[KNN - 1] = m0 + ml;
#pragma unroll
                    for (int j = KNN - 1; j > 0; --j) {
                        if (tv[j] < tv[j - 1]) {
                            float tf = tv[j]; tv[j] = tv[j - 1]; tv[j - 1] = tf;
                            int   tx = ti[j]; ti[j] = ti[j - 1]; ti[j - 1] = tx;
                        }
                    }
                }
            }
        }
    }
    if (tid < TILE) {
        int* orow = knn + ((size_t)b * NTOK + n0 + tid) * KNN;
#pragma unroll
        for (int j = 0; j < KNN; ++j) orow[j] = ti[j];
    }
}

// ------------------------------- 128x128 FP32-WMMA GEMM tile helper
// A, B, D all in interleaved-pair layout (ilx). D[n][m] so it can serve as
// the next layer's A with zero reformatting.
__device__ __forceinline__ void gemm128(const float* __restrict__ A,
                                        const float* __restrict__ Bm,
                                        const float* __restrict__ bias,
                                        bool relu, float* __restrict__ D,
                                        int tid) {
    const int lane = tid & 31, wv = tid >> 5, half = lane >> 4, l16 = lane & 15;
    v8f acc[8];
#pragma unroll
    for (int ni = 0; ni < 8; ++ni) acc[ni] = (v8f){0,0,0,0,0,0,0,0};
    const int mrow = wv * 16 + l16;
    for (int c0 = 0; c0 < C; c0 += 4) {
        const int kp = (c0 >> 1) + half;
        v2f a = *(const v2f*)&A[kp * SP + mrow * 2];
#pragma unroll
        for (int ni = 0; ni < 8; ++ni) {
            v2f bf = *(const v2f*)&Bm[kp * SP + (ni * 16 + l16) * 2];
            acc[ni] = __builtin_amdgcn_wmma_f32_16x16x4_f32(
                false, a, false, bf, (short)0, acc[ni], false, false);
        }
    }
#pragma unroll
    for (int ni = 0; ni < 8; ++ni) {
        int col = ni * 16 + l16;                   // output channel d
        float bv = bias[col];
#pragma unroll
        for (int r = 0; r < 8; ++r) {
            int row = wv * 16 + half * 8 + r;      // token index
            float v = acc[ni][r] + bv;
            if (relu) v = fmaxf(v, 0.0f);
            D[ilx(col, row)] = v;
        }
    }
}

// ------------------------- gather + MLP (two WMMA GEMMs) + BN stats
__global__ __launch_bounds__(256) void k_mlp(const float* __restrict__ x,
                                             const float* __restrict__ eps_p,
                                             const float* __restrict__ W1,
                                             const float* __restrict__ b1,
                                             const float* __restrict__ W2,
                                             const float* __restrict__ b2,
                                             const int* __restrict__ knn,
                                             float* __restrict__ y,
                                             float* __restrict__ sums,
                                             float* __restrict__ sumsq) {
    extern __shared__ char smem[];
    float* lds_h  = (float*)smem;                  // H0, later Y2 (interleaved)
    float* lds_h2 = lds_h + (C / 2) * SP;          // H1 (interleaved)
    float* lds_w  = lds_h2 + (C / 2) * SP;         // W1 then W2 (interleaved)
    int*   lds_i  = (int*)(lds_w + (C / 2) * SP);  // [TILE*KNN]

    const int tid = threadIdx.x;
    const int b   = blockIdx.x / NB;
    const int n0  = (blockIdx.x % NB) * TILE;
    const float* xb = x + (size_t)b * C * NTOK;

    for (int i = tid; i < TILE * KNN; i += 256)
        lds_i[i] = knn[((size_t)b * NTOK + n0) * KNN + i];
    __syncthreads();

    const float epsw = 1.0f + eps_p[0];
    // H0[t][c] = (1+eps)*x[c][n0+t] + sum_j x[c][idx_j]
    for (int i = tid; i < C * TILE; i += 256) {
        int c = i >> 7, t = i & 127;
        const float* xc = xb + (size_t)c * NTOK;
        float s = epsw * xc[n0 + t];
#pragma unroll
        for (int j = 0; j < KNN; ++j) s += xc[lds_i[t * KNN + j]];
        lds_h[ilx(c, t)] = s;
    }
    for (int i = tid; i < C * C; i += 256) {
        int c = i >> 7, d = i & 127;
        lds_w[ilx(c, d)] = W1[i];
    }
    __syncthreads();

    gemm128(lds_h, lds_w, b1, true, lds_h2, tid);     // layer 1 + ReLU
    __syncthreads();
    for (int i = tid; i < C * C; i += 256) {
        int c = i >> 7, d = i & 127;
        lds_w[ilx(c, d)] = W2[i];
    }
    __syncthreads();
    gemm128(lds_h2, lds_w, b2, false, lds_h, tid);    // layer 2
    __syncthreads();

    // coalesced store of pre-BN output [b][d][n]
    for (int i = tid; i < C * TILE; i += 256) {
        int d = i >> 7, t = i & 127;
        y[((size_t)b * C + d) * NTOK + n0 + t] = lds_h[ilx(d, t)];
    }
    // per-channel partial sums for batchnorm
    if (tid < C) {
        float s = 0.0f, s2 = 0.0f;
        for (int t = 0; t < TILE; ++t) {
            float v = lds_h[ilx(tid, t)];
            s += v; s2 += v * v;
        }
        atomicAdd(&sums[tid], s);
        atomicAdd(&sumsq[tid], s2);
    }
}

// ---------------------------------------- BN + residual + ReLU
__global__ __launch_bounds__(256) void k_bn(const float* __restrict__ y,
                                            const float* __restrict__ x,
                                            const float* __restrict__ gamma,
                                            const float* __restrict__ beta,
                                            const float* __restrict__ sums,
                                            const float* __restrict__ sumsq,
                                            float* __restrict__ out) {
    size_t gid = (size_t)blockIdx.x * 256 + threadIdx.x;  // over B*C*N
    int c = (int)((gid / NTOK) % C);
    const float inv = 1.0f / (float)(BB * NTOK);
    float mean = sums[c] * inv;
    float var  = sumsq[c] * inv - mean * mean;
    float v    = y[gid];
    float bn   = gamma[c] * (v - mean) * rsqrtf(var + 1e-5f) + beta[c];
    float o    = bn + x[gid];
    out[gid]   = fmaxf(o, 0.0f);
}

// ---------------------------------------------------------------- host
extern "C" void kernel_launch(void* const* d_in, const int* in_sizes, int n_in,
                              void* d_out, int out_size, void* d_ws, size_t ws_size,
                              hipStream_t stream) {
    (void)in_sizes; (void)n_in; (void)out_size; (void)ws_size;
    const float* x     = (const float*)d_in[0];
    const float* eps_p = (const float*)d_in[1];
    const float* W1    = (const float*)d_in[2];
    const float* b1    = (const float*)d_in[3];
    const float* W2    = (const float*)d_in[4];
    const float* b2    = (const float*)d_in[5];
    const float* gamma = (const float*)d_in[6];
    const float* beta  = (const float*)d_in[7];
    float* out = (float*)d_out;

    char* ws = (char*)d_ws;
    float* sq    = (float*)ws;                     // 4*4096 f        = 64 KB
    int*   knn   = (int*)(ws + 65536);             // 4*4096*9 i      = 576 KB
    float* y     = (float*)(ws + 655360);          // 4*128*4096 f    = 8 MB
    float* sums  = (float*)(ws + 9043968);         // 128 f
    float* sumsq = (float*)(ws + 9044480);         // 128 f

    k_init<<<1, 256, 0, stream>>>(sums, sumsq);
    k_sqnorm<<<(BB * NTOK) / 256, 256, 0, stream>>>(x, sq);

    size_t sh1 = (size_t)(2 * (C / 2) * SP + TILE * LDA + 2 * TILE) * sizeof(float);
    k_gram_topk<<<BB * NB, 256, sh1, stream>>>(x, sq, knn);

    size_t sh2 = (size_t)(3 * (C / 2) * SP) * sizeof(float) + TILE * KNN * sizeof(int);
    k_mlp<<<BB * NB, 256, sh2, stream>>>(x, eps_p, W1, b1, W2, b2, knn, y, sums, sumsq);

    k_bn<<<(BB * C * NTOK) / 256, 256, 0, stream>>>(y, x, gamma, beta, sums, sumsq, out);
}